// Model_6871947673994
// MI455X (gfx1250) — compile-verified
//
#include <hip/hip_runtime.h>

typedef float v2f __attribute__((ext_vector_type(2)));
typedef float v8f __attribute__((ext_vector_type(8)));

#define NB    1024
#define DIN   512
#define DOUT  640
#define NCLS  1000
#define KMAX  192
#define EPSF  1e-9f
#define NT    4      // N-tiles per wave (16x64 output strip per wave)

// -------- f32 WMMA helper: D = A(16x4) * B(4x16) + C, full fp32 --------
static __device__ __forceinline__ v8f wmma4(v2f a, v2f b, v8f c) {
  // 8 args: (neg_a, A, neg_b, B, c_mod, C, reuse_a, reuse_b)
  return __builtin_amdgcn_wmma_f32_16x16x4_f32(
      false, a, false, b, (short)0, c, false, false);
}

// contiguous 8B fragment load (k-pairs are 8B aligned: k0 % 4 == 0, khalf even)
static __device__ __forceinline__ v2f ld2(const float* __restrict__ p) {
  return *reinterpret_cast<const v2f*>(p);
}

// ---------------- AtA = A^T @ A  (DIN x DIN), K = DOUT ----------------
// strided A-side loads (column gathers); wave computes 16 x 64 strip
__global__ void k_ata(const float* __restrict__ A, float* __restrict__ AtA) {
  const int lane  = threadIdx.x;
  const int mR    = blockIdx.x * 16;
  const int nC    = blockIdx.y * (16 * NT);
  const int mrow  = mR + (lane & 15);
  const int nlane = lane & 15;
  const int khalf = (lane >> 4) << 1;
  v8f acc[NT] = {};
  for (int k0 = 0; k0 < DOUT; k0 += 4) {
    const int kk = k0 + khalf;
    v2f a;
    a.x = A[kk * DIN + mrow];
    a.y = A[(kk + 1) * DIN + mrow];
#pragma unroll
    for (int t = 0; t < NT; ++t) {
      const int ncol = nC + t * 16 + nlane;
      v2f b;
      b.x = A[kk * DIN + ncol];
      b.y = A[(kk + 1) * DIN + ncol];
      acc[t] = wmma4(a, b, acc[t]);
    }
  }
#pragma unroll
  for (int t = 0; t < NT; ++t) {
    const int ncol = nC + t * 16 + nlane;
#pragma unroll
    for (int v = 0; v < 8; ++v) {
      const int m = mR + v + ((lane >> 4) << 3);
      AtA[m * DIN + ncol] = acc[t][v];
    }
  }
}

// ------------- Z = (f .* m) @ A^T  (NB x DOUT), K = DIN ---------------
__global__ void k_z(const float* __restrict__ f,
                    const unsigned char* __restrict__ mask,
                    const float* __restrict__ A, float* __restrict__ Z) {
  const int lane  = threadIdx.x;
  const int mR    = blockIdx.x * 16;
  const int nC    = blockIdx.y * (16 * NT);
  const int mrow  = mR + (lane & 15);
  const int nlane = lane & 15;
  const int khalf = (lane >> 4) << 1;
  v8f acc[NT] = {};
  for (int k0 = 0; k0 < DIN; k0 += 4) {
    const int kk = k0 + khalf;
    v2f a = ld2(f + mrow * DIN + kk);               // b64 load
    a.x *= (float)mask[mrow * DIN + kk];
    a.y *= (float)mask[mrow * DIN + kk + 1];
#pragma unroll
    for (int t = 0; t < NT; ++t) {
      const int ncol = nC + t * 16 + nlane;
      const v2f b = ld2(A + ncol * DIN + kk);       // b64 load
      acc[t] = wmma4(a, b, acc[t]);
    }
  }
#pragma unroll
  for (int t = 0; t < NT; ++t) {
    const int ncol = nC + t * 16 + nlane;
#pragma unroll
    for (int v = 0; v < 8; ++v) {
      const int m = mR + v + ((lane >> 4) << 3);
      Z[m * DOUT + ncol] = acc[t][v];
    }
  }
}

// ------------- R = m .* (Z @ A)  (NB x DIN), K = DOUT -----------------
__global__ void k_rhs(const float* __restrict__ Z,
                      const unsigned char* __restrict__ mask,
                      const float* __restrict__ A, float* __restrict__ R) {
  const int lane  = threadIdx.x;
  const int mR    = blockIdx.x * 16;
  const int nC    = blockIdx.y * (16 * NT);
  const int mrow  = mR + (lane & 15);
  const int nlane = lane & 15;
  const int khalf = (lane >> 4) << 1;
  v8f acc[NT] = {};
  for (int k0 = 0; k0 < DOUT; k0 += 4) {
    const int kk = k0 + khalf;
    const v2f a = ld2(Z + mrow * DOUT + kk);        // b64 load
#pragma unroll
    for (int t = 0; t < NT; ++t) {
      const int ncol = nC + t * 16 + nlane;
      v2f b;
      b.x = A[kk * DIN + ncol];
      b.y = A[(kk + 1) * DIN + ncol];
      acc[t] = wmma4(a, b, acc[t]);
    }
  }
#pragma unroll
  for (int t = 0; t < NT; ++t) {
    const int ncol = nC + t * 16 + nlane;
#pragma unroll
    for (int v = 0; v < 8; ++v) {
      const int m = mR + v + ((lane >> 4) << 3);
      R[m * DIN + ncol] = mask[m * DIN + ncol] ? acc[t][v] : 0.0f;
    }
  }
}

// ------ per-row gather + Cholesky solve on erased block (LDS) ---------
// One 256-thread workgroup per batch row. Dynamic LDS:
//   sG[KMAX*KMAX] | sV[KMAX] | sIdx[KMAX]
__launch_bounds__(256)
__global__ void k_solve(const float* __restrict__ f,
                        const unsigned char* __restrict__ mask,
                        const float* __restrict__ AtA,
                        const float* __restrict__ R,
                        float* __restrict__ xrec,
                        float* __restrict__ mse_out,
                        float* __restrict__ fracc_out,
                        float* __restrict__ ner_out) {
  extern __shared__ float smem[];
  float* sG   = smem;                       // KMAX*KMAX
  float* sV   = smem + KMAX * KMAX;         // KMAX
  int*   sIdx = (int*)(sV + KMAX);          // KMAX
  __shared__ float sRed[256];
  __shared__ float sRed2[256];
  __shared__ float sRed3[256];
  __shared__ int   sCnt;

  const int row = blockIdx.x;
  const int tid = threadIdx.x;

  // deterministic gather of erased column indices
  if (tid == 0) {
    int c = 0;
    for (int d = 0; d < DIN; ++d) {
      if (mask[row * DIN + d]) {
        if (c < KMAX) sIdx[c] = d;
        ++c;
      }
    }
    sCnt = c;
  }
  __syncthreads();
  const int ne = sCnt;
  const int K  = ne < KMAX ? ne : KMAX;

  // G_E = AtA[idx,idx] gathered out of L2 ; rhs_E = R[row, idx]
  for (int t = tid; t < K * K; t += 256) {
    const int i = t / K, j = t % K;
    sG[t] = AtA[sIdx[i] * DIN + sIdx[j]];
  }
  for (int t = tid; t < K; t += 256) sV[t] = R[row * DIN + sIdx[t]];
  __syncthreads();

  // in-place Cholesky (lower)
  for (int j = 0; j < K; ++j) {
    if (tid == 0) sG[j * K + j] = sqrtf(sG[j * K + j]);
    __syncthreads();
    const float dj = sG[j * K + j];
    for (int i = j + 1 + tid; i < K; i += 256) sG[i * K + j] /= dj;
    __syncthreads();
    const int rem = K - j - 1;
    for (int t = tid; t < rem * rem; t += 256) {
      const int r = j + 1 + t / rem;
      const int c = j + 1 + t % rem;
      if (c <= r) sG[r * K + c] -= sG[r * K + j] * sG[c * K + j];
    }
    __syncthreads();
  }

  // forward solve L y = rhs
  for (int j = 0; j < K; ++j) {
    if (tid == 0) sV[j] /= sG[j * K + j];
    __syncthreads();
    const float yj = sV[j];
    for (int i = j + 1 + tid; i < K; i += 256) sV[i] -= sG[i * K + j] * yj;
    __syncthreads();
  }
  // backward solve L^T x = y
  for (int j = K - 1; j >= 0; --j) {
    if (tid == 0) sV[j] /= sG[j * K + j];
    __syncthreads();
    const float xj = sV[j];
    for (int i = tid; i < j; i += 256) sV[i] -= sG[j * K + i] * xj;
    __syncthreads();
  }

  // x_rec = where(mask, sol, f)
  for (int d = tid; d < DIN; d += 256) xrec[row * DIN + d] = f[row * DIN + d];
  __syncthreads();
  for (int t = tid; t < K; t += 256) xrec[row * DIN + sIdx[t]] = sV[t];

  // mse over erased positions
  float part = 0.0f;
  for (int t = tid; t < K; t += 256) {
    const float dlt = sV[t] - f[row * DIN + sIdx[t]];
    part += dlt * dlt;
  }
  sRed[tid] = part;
  // row variance (population): sums of f and f^2
  float s1 = 0.0f, s2 = 0.0f;
  for (int d = tid; d < DIN; d += 256) {
    const float x = f[row * DIN + d];
    s1 += x;
    s2 += x * x;
  }
  sRed2[tid] = s1;
  sRed3[tid] = s2;
  __syncthreads();
  for (int s = 128; s > 0; s >>= 1) {
    if (tid < s) {
      sRed[tid]  += sRed[tid + s];
      sRed2[tid] += sRed2[tid + s];
      sRed3[tid] += sRed3[tid + s];
    }
    __syncthreads();
  }
  if (tid == 0) {
    const float nef  = (float)ne;
    const float mse  = sRed[0] / fmaxf(nef, 1.0f);
    const float mean = sRed2[0] / (float)DIN;
    const float var  = sRed3[0] / (float)DIN - mean * mean;
    const float rel  = sqrtf(mse + EPSF) / sqrtf(var + EPSF);
    mse_out[row]   = mse;
    fracc_out[row] = fminf(fmaxf(1.0f - rel, 0.0f), 1.0f);
    ner_out[row]   = nef;
  }
}

// ---------- logits = x_rec @ W^T + b  (NB x NCLS), K = DIN ------------
__global__ void k_logits(const float* __restrict__ xrec,
                         const float* __restrict__ W,
                         const float* __restrict__ bias,
                         float* __restrict__ out) {
  const int lane  = threadIdx.x;
  const int mR    = blockIdx.x * 16;
  const int nC    = blockIdx.y * (16 * NT);
  const int mrow  = mR + (lane & 15);
  const int nlane = lane & 15;
  const int khalf = (lane >> 4) << 1;
  int   nsafe[NT];
  float nmul[NT];
#pragma unroll
  for (int t = 0; t < NT; ++t) {
    const int ncol = nC + t * 16 + nlane;
    nsafe[t] = ncol < NCLS ? ncol : (NCLS - 1);
    nmul[t]  = ncol < NCLS ? 1.0f : 0.0f;  // branchless pad, EXEC stays full
  }
  v8f acc[NT] = {};
  for (int k0 = 0; k0 < DIN; k0 += 4) {
    const int kk = k0 + khalf;
    const v2f a = ld2(xrec + mrow * DIN + kk);      // b64 load
#pragma unroll
    for (int t = 0; t < NT; ++t) {
      v2f b = ld2(W + nsafe[t] * DIN + kk);         // b64 load, in-bounds
      b.x *= nmul[t];
      b.y *= nmul[t];
      acc[t] = wmma4(a, b, acc[t]);
    }
  }
#pragma unroll
  for (int t = 0; t < NT; ++t) {
    const int ncol = nC + t * 16 + nlane;
    if (ncol < NCLS) {
      const float bv = bias[ncol];
#pragma unroll
      for (int v = 0; v < 8; ++v) {
        const int m = mR + v + ((lane >> 4) << 3);
        out[m * NCLS + ncol] = acc[t][v] + bv;
      }
    }
  }
}

extern "C" void kernel_launch(void* const* d_in, const int* in_sizes, int n_in,
                              void* d_out, int out_size, void* d_ws, size_t ws_size,
                              hipStream_t stream) {
  const float*         f    = (const float*)d_in[0];
  const unsigned char* mask = (const unsigned char*)d_in[1];  // jnp bool -> 1 byte
  const float*         A    = (const float*)d_in[2];
  const float*         W    = (const float*)d_in[3];
  const float*         bias = (const float*)d_in[4];

  float* out    = (float*)d_out;
  float* logits = out;                    // NB*NCLS
  float* mse    = out + NB * NCLS;        // NB
  float* fracc  = mse + NB;               // NB
  float* ner    = fracc + NB;             // NB

  float* ws  = (float*)d_ws;
  float* AtA = ws;                        // DIN*DIN
  float* Z   = AtA + DIN * DIN;           // NB*DOUT
  float* R   = Z + NB * DOUT;             // NB*DIN
  float* XR  = R + NB * DIN;              // NB*DIN

  const dim3 wblk(32);  // one wave32 per block: 16 x (16*NT) WMMA strip

  k_ata   <<<dim3(DIN / 16, DIN / (16 * NT)),  wblk, 0, stream>>>(A, AtA);
  k_z     <<<dim3(NB / 16, DOUT / (16 * NT)),  wblk, 0, stream>>>(f, mask, A, Z);
  k_rhs   <<<dim3(NB / 16, DIN / (16 * NT)),   wblk, 0, stream>>>(Z, mask, A, R);

  const size_t solve_lds = (size_t)(KMAX * KMAX + 2 * KMAX) * sizeof(float);
  k_solve <<<dim3(NB), dim3(256), solve_lds, stream>>>(f, mask, AtA, R, XR,
                                                       mse, fracc, ner);

  k_logits<<<dim3(NB / 16, (NCLS + 16 * NT - 1) / (16 * NT)), wblk, 0, stream>>>(
      XR, W, bias, logits);
}